// GNN_60559038874107
// MI455X (gfx1250) — compile-verified
//
#include <hip/hip_runtime.h>

#define N_NODES 20000
#define N_EDGES 160000
#define D 512

typedef __attribute__((ext_vector_type(2))) float v2f;
typedef __attribute__((ext_vector_type(8))) float v8f;

// ---------------------------------------------------------------------------
// Fold w_root + w_lin into one weight, b_rel + b_lin into one bias.
// ---------------------------------------------------------------------------
__global__ __launch_bounds__(256) void combine_wb_kernel(
    const float* __restrict__ wroot, const float* __restrict__ wlin,
    const float* __restrict__ brel,  const float* __restrict__ blin,
    float* __restrict__ wcomb, float* __restrict__ bcomb) {
  int i = blockIdx.x * 256 + threadIdx.x;
  if (i < D * D) wcomb[i] = wroot[i] + wlin[i];
  if (i < D)     bcomb[i] = brel[i] + blin[i];
}

// ---------------------------------------------------------------------------
// Edge scatter-add: agg[dst] += x[src].  128 threads per edge, float4 per
// thread, f32 global atomics (L2-resident: x is 41 MB < 192 MB L2).
// ---------------------------------------------------------------------------
__global__ __launch_bounds__(256) void scatter_add_kernel(
    const float* __restrict__ x, const long long* __restrict__ ei,
    float* __restrict__ agg) {
  unsigned tid = blockIdx.x * 256u + threadIdx.x;
  unsigned e = tid >> 7;           // edge index
  unsigned c = (tid & 127u) << 2;  // feature offset (4 floats)
  if (e >= N_EDGES) return;
  int s = (int)ei[e];              // edge_index[0][e]
  int d = (int)ei[N_EDGES + e];    // edge_index[1][e]
  const float4 v = *(const float4*)(x + (size_t)s * D + c);
  float* dstp = agg + (size_t)d * D + c;
  atomicAdd(dstp + 0, v.x);
  atomicAdd(dstp + 1, v.y);
  atomicAdd(dstp + 2, v.z);
  atomicAdd(dstp + 3, v.w);
}

// ---------------------------------------------------------------------------
// Fused dual GEMM: out = act( agg @ wrel^T + xin @ wcomb^T + bias )
// One wave computes a 16(M) x 64(N) tile with 4 f32 16x16 accumulators,
// accumulating over K=512 with exact-fp32 v_wmma_f32_16x16x4_f32.
// A-frag (16x4 f32): lane(0-15)=row, vgpr pair = K{0,1}; lanes 16-31 = K{2,3}.
// B-frag (4x16 f32): same per-lane addressing with "row" = weight output n
// (weights are row-major [n][k], i.e. already B = W^T column tiles).
// ---------------------------------------------------------------------------
__global__ __launch_bounds__(256) void gemm_fused_kernel(
    const float* __restrict__ agg, const float* __restrict__ xin,
    const float* __restrict__ wrel, const float* __restrict__ wcomb,
    const float* __restrict__ bias, float* __restrict__ out, int relu) {
  const int lane = threadIdx.x & 31;
  const int wave = threadIdx.x >> 5;
  const int tile = blockIdx.x * 8 + wave;   // 1250*8 = 10000 wave-tiles
  const int m0 = (tile >> 3) << 4;          // 0..19984
  const int n0 = (tile & 7) << 6;           // 0..448
  const int r = lane & 15;
  const int khalf = (lane >> 4) << 1;       // 0 or 2

  const float* pa = agg  + (size_t)(m0 + r) * D + khalf;
  const float* px = xin  + (size_t)(m0 + r) * D + khalf;
  const float* pr = wrel + (size_t)(n0 + r) * D + khalf;
  const float* pc = wcomb+ (size_t)(n0 + r) * D + khalf;

  v8f acc0 = {}, acc1 = {}, acc2 = {}, acc3 = {};

#pragma unroll 4
  for (int k = 0; k < D; k += 4) {
    v2f aa  = *(const v2f*)(pa + k);
    v2f ax  = *(const v2f*)(px + k);
    v2f br0 = *(const v2f*)(pr + k);
    v2f br1 = *(const v2f*)(pr + 16 * D + k);
    v2f br2 = *(const v2f*)(pr + 32 * D + k);
    v2f br3 = *(const v2f*)(pr + 48 * D + k);
    v2f bc0 = *(const v2f*)(pc + k);
    v2f bc1 = *(const v2f*)(pc + 16 * D + k);
    v2f bc2 = *(const v2f*)(pc + 32 * D + k);
    v2f bc3 = *(const v2f*)(pc + 48 * D + k);
    acc0 = __builtin_amdgcn_wmma_f32_16x16x4_f32(false, aa, false, br0, (short)0, acc0, false, false);
    acc1 = __builtin_amdgcn_wmma_f32_16x16x4_f32(false, aa, false, br1, (short)0, acc1, false, false);
    acc2 = __builtin_amdgcn_wmma_f32_16x16x4_f32(false, aa, false, br2, (short)0, acc2, false, false);
    acc3 = __builtin_amdgcn_wmma_f32_16x16x4_f32(false, aa, false, br3, (short)0, acc3, false, false);
    acc0 = __builtin_amdgcn_wmma_f32_16x16x4_f32(false, ax, false, bc0, (short)0, acc0, false, false);
    acc1 = __builtin_amdgcn_wmma_f32_16x16x4_f32(false, ax, false, bc1, (short)0, acc1, false, false);
    acc2 = __builtin_amdgcn_wmma_f32_16x16x4_f32(false, ax, false, bc2, (short)0, acc2, false, false);
    acc3 = __builtin_amdgcn_wmma_f32_16x16x4_f32(false, ax, false, bc3, (short)0, acc3, false, false);
  }

  // C/D layout: vgpr i -> M = i + 8*(lane>=16), N = n0 + 16*j + (lane&15)
  const int rowHi = (lane >> 4) << 3;
  const int n = n0 + r;
  const float b0 = bias[n], b1 = bias[n + 16], b2 = bias[n + 32], b3 = bias[n + 48];
#pragma unroll
  for (int i = 0; i < 8; i++) {
    float v0 = acc0[i] + b0;
    float v1 = acc1[i] + b1;
    float v2 = acc2[i] + b2;
    float v3 = acc3[i] + b3;
    if (relu) {
      v0 = fmaxf(v0, 0.f); v1 = fmaxf(v1, 0.f);
      v2 = fmaxf(v2, 0.f); v3 = fmaxf(v3, 0.f);
    }
    float* o = out + (size_t)(m0 + rowHi + i) * D;
    o[n]      = v0;
    o[n + 16] = v1;
    o[n + 32] = v2;
    o[n + 48] = v3;
  }
}

// ---------------------------------------------------------------------------
extern "C" void kernel_launch(void* const* d_in, const int* in_sizes, int n_in,
                              void* d_out, int out_size, void* d_ws, size_t ws_size,
                              hipStream_t stream) {
  (void)in_sizes; (void)n_in; (void)out_size; (void)ws_size;

  const float*     x  = (const float*)d_in[0];
  const long long* ei = (const long long*)d_in[1];  // int64 edge_index [2, E]

  const float* w_rel[3]  = { (const float*)d_in[2],  (const float*)d_in[7],  (const float*)d_in[12] };
  const float* b_rel[3]  = { (const float*)d_in[3],  (const float*)d_in[8],  (const float*)d_in[13] };
  const float* w_root[3] = { (const float*)d_in[4],  (const float*)d_in[9],  (const float*)d_in[14] };
  const float* w_lin[3]  = { (const float*)d_in[5],  (const float*)d_in[10], (const float*)d_in[15] };
  const float* b_lin[3]  = { (const float*)d_in[6],  (const float*)d_in[11], (const float*)d_in[16] };

  const size_t FEAT_BYTES = (size_t)N_NODES * D * sizeof(float);  // 40.96 MB
  char* ws  = (char*)d_ws;
  float* agg = (float*)ws;                              // [N, D]
  float* hid = (float*)(ws + FEAT_BYTES);               // [N, D] ping buffer
  float* wc  = (float*)(ws + 2 * FEAT_BYTES);           // 3 x [D, D]
  float* bc  = (float*)(ws + 2 * FEAT_BYTES + 3 * (size_t)D * D * sizeof(float)); // 3 x [D]

  // Fold the two node-side linears of each layer into one weight/bias.
  for (int l = 0; l < 3; l++) {
    combine_wb_kernel<<<(D * D + 255) / 256, 256, 0, stream>>>(
        w_root[l], w_lin[l], b_rel[l], b_lin[l],
        wc + (size_t)l * D * D, bc + (size_t)l * D);
  }

  // Ping-pong: x -> d_out -> hid -> d_out  (d_out reused as scratch; only its
  // final contents are validated).
  const float* cur = x;
  float* outs[3] = { (float*)d_out, hid, (float*)d_out };
  for (int l = 0; l < 3; l++) {
    hipMemsetAsync(agg, 0, FEAT_BYTES, stream);
    scatter_add_kernel<<<(N_EDGES * 128) / 256, 256, 0, stream>>>(cur, ei, agg);
    gemm_fused_kernel<<<N_NODES / 16, 256, 0, stream>>>(
        agg, cur, w_rel[l], wc + (size_t)l * D * D, bc + (size_t)l * D,
        outs[l], (l < 2) ? 1 : 0);
    cur = outs[l];
  }
}